// SentimentNetEnd2End_27685359190258
// MI455X (gfx1250) — compile-verified
//
#include <hip/hip_runtime.h>
#include <stdint.h>

// ---------------------------------------------------------------------------
// Fused LSTM(T=512, I=H=3) + cosine-similarity kernel for gfx1250 (MI455X).
//   - thread-per-sequence, wave32, one wave per workgroup (512 waves total)
//   - story tiles staged Global->LDS by the Tensor Data Mover (TDM),
//     double-buffered on TENSORcnt; TDM pad feature gives a 99-dword row
//     stride => conflict-free LDS banks (gcd(99,64)=1)
//   - activations via v_tanh_f32; sigmoid gates (i,f,o) have the x/2 folded
//     into the (wave-uniform, SGPR-resident) weights at load time, so
//     sigmoid costs exactly one v_tanh + one v_fma per gate.
// ---------------------------------------------------------------------------

#define TC        32                    // timesteps per staged chunk
#define ROW_DW    (TC * 3)              // 96 dwords of payload per (row, chunk)
#define LROW_DW   (ROW_DW + ROW_DW/32)  // 99: +1 pad dword per 32 stored dwords
#define BUF_DW    (32 * LROW_DW)        // 3168 dwords = 12672 B per buffer
#define BUF_BYTES (BUF_DW * 4)

typedef __attribute__((ext_vector_type(4))) uint32_t u32x4;
typedef __attribute__((ext_vector_type(8))) uint32_t u32x8;

__device__ __forceinline__ float fast_tanh(float x) {
#if __has_builtin(__builtin_amdgcn_tanhf)
    return __builtin_amdgcn_tanhf(x);                          // v_tanh_f32
#else
    float e = __builtin_amdgcn_exp2f(x * 2.8853900817779268f); // exp(2x)
    return 1.0f - 2.0f * __builtin_amdgcn_rcpf(e + 1.0f);
#endif
}
// Input already pre-scaled by 1/2 via folded weights: sigmoid(2y)|y=x/2
__device__ __forceinline__ float sigm_prescaled(float y) {
    return fmaf(fast_tanh(y), 0.5f, 0.5f);
}
__device__ __forceinline__ float fast_rcp(float x) {
#if __has_builtin(__builtin_amdgcn_rcpf)
    return __builtin_amdgcn_rcpf(x);
#else
    return 1.0f / x;
#endif
}

struct LstmState { float h0, h1, h2, c0, c1, c2; };

// wih/whh/bias for gates 0-5 (i,f) and 9-11 (o) are pre-scaled by 0.5.
__device__ __forceinline__ void lstm_step(float x0, float x1, float x2,
                                          const float (&wih)[12][3],
                                          const float (&whh)[12][3],
                                          const float (&bias)[12],
                                          LstmState& s) {
    float p[12];
#pragma unroll
    for (int g = 0; g < 12; ++g) {
        p[g] = fmaf(x0, wih[g][0],
               fmaf(x1, wih[g][1],
               fmaf(x2, wih[g][2],
               fmaf(s.h0, whh[g][0],
               fmaf(s.h1, whh[g][1],
               fmaf(s.h2, whh[g][2], bias[g]))))));
    }
    // PyTorch gate order: i, f, g, o
    float i0 = sigm_prescaled(p[0]),  i1 = sigm_prescaled(p[1]),  i2 = sigm_prescaled(p[2]);
    float f0 = sigm_prescaled(p[3]),  f1 = sigm_prescaled(p[4]),  f2 = sigm_prescaled(p[5]);
    float g0 = fast_tanh(p[6]),       g1 = fast_tanh(p[7]),       g2 = fast_tanh(p[8]);
    float o0 = sigm_prescaled(p[9]),  o1 = sigm_prescaled(p[10]), o2 = sigm_prescaled(p[11]);
    s.c0 = fmaf(f0, s.c0, i0 * g0);
    s.c1 = fmaf(f1, s.c1, i1 * g1);
    s.c2 = fmaf(f2, s.c2, i2 * g2);
    s.h0 = o0 * fast_tanh(s.c0);
    s.h1 = o1 * fast_tanh(s.c1);
    s.h2 = o2 * fast_tanh(s.c2);
}

// Issue one TDM 2-D tile load: 32 rows x ROW_DW dwords, global row stride =
// row_len elems, with LDS padding of 1 dword per 32 dwords (pad_interval=4,
// pad_amount=0) so each row lands at a 99-dword stride in LDS.
__device__ __forceinline__ void tdm_issue_tile(uint32_t lds_byte, uint64_t gaddr,
                                               uint32_t row_len /* 3*T elems */) {
    u32x4 g0;
    g0[0] = 1u;                                              // count=1, user mode
    g0[1] = lds_byte;                                        // LDS byte address
    g0[2] = (uint32_t)gaddr;                                 // global addr lo
    g0[3] = (uint32_t)((gaddr >> 32) & 0x01FFFFFFu) | (2u << 30); // hi | type=2
    u32x8 g1;
    g1[0] = (2u << 16)      // data_size = 4 B
          | (1u << 20)      // pad_enable
          | (4u << 22)      // pad_interval: 4 -> every 32 dwords
          | (0u << 25);     // pad_amount:   0 -> 1 dword
    g1[1] = (row_len & 0xFFFFu) << 16;               // tensor_dim0 lo16
    g1[2] = (row_len >> 16);                         // tensor_dim0 hi | dim1 lo (dim1=1<<20)
    g1[3] = 0x10u | ((uint32_t)ROW_DW << 16);        // dim1 hi(0x10) | tile_dim0=96
    g1[4] = 32u;                                     // tile_dim1=32 rows, tile_dim2=0
    g1[5] = row_len;                                 // tensor_dim0_stride lo (elems)
    g1[6] = 0u;                                      // stride0 hi | stride1 lo
    g1[7] = 0u;
    asm volatile("tensor_load_to_lds %0, %1" :: "s"(g0), "s"(g1) : "memory");
}

__global__ __launch_bounds__(32)
void lstm_cosine_kernel(const float* __restrict__ story,
                        const float* __restrict__ e1,
                        const float* __restrict__ e2,
                        const float* __restrict__ Wih,
                        const float* __restrict__ Whh,
                        const float* __restrict__ bih,
                        const float* __restrict__ bhh,
                        float* __restrict__ out,
                        int B, int T) {
    __shared__ float lds[2 * BUF_DW];                 // 25,344 B of 320 KB

    const int lane = threadIdx.x;                     // wave32: lane == tid
    const int b0   = blockIdx.x * 32;
    const int b    = b0 + lane;

    // Wave-uniform weights -> scalar loads -> SGPRs. Fold the sigmoid 1/2
    // pre-scale into weights+bias of gates i (0-2), f (3-5), o (9-11).
    float wih[12][3], whh[12][3], bias[12];
#pragma unroll
    for (int k = 0; k < 12; ++k) {
        const float sc = (k >= 6 && k <= 8) ? 1.0f : 0.5f;   // g-gate unscaled
#pragma unroll
        for (int j = 0; j < 3; ++j) {
            wih[k][j] = Wih[k * 3 + j] * sc;
            whh[k][j] = Whh[k * 3 + j] * sc;
        }
        bias[k] = (bih[k] + bhh[k]) * sc;
    }

    LstmState s = {0.f, 0.f, 0.f, 0.f, 0.f, 0.f};

    const int full_tile = (b0 + 32 <= B);
    const int chunks    = full_tile ? (T / TC) : 0;
    const uint32_t row_len = 3u * (uint32_t)T;

    if (chunks > 0) {
        uint32_t lds_base;
        {   // LDS byte offset of `lds` (ptrtoint in addrspace(3))
            __attribute__((address_space(3))) float* p3 =
                (__attribute__((address_space(3))) float*)lds;
            lds_base = (uint32_t)(uintptr_t)p3;
        }
        const uint64_t gtile =
            (uint64_t)(uintptr_t)story + 4ull * (uint64_t)b0 * (uint64_t)row_len;

        tdm_issue_tile(lds_base, gtile, row_len);     // prefetch chunk 0
        for (int k = 0; k < chunks; ++k) {
            if (k + 1 < chunks) {
                tdm_issue_tile(lds_base + (uint32_t)(((k + 1) & 1) * BUF_BYTES),
                               gtile + 4ull * (uint64_t)(3 * TC) * (uint64_t)(k + 1),
                               row_len);
                asm volatile("s_wait_tensorcnt 0x1" ::: "memory"); // chunk k ready
            } else {
                asm volatile("s_wait_tensorcnt 0x0" ::: "memory");
            }
            const int cb = (k & 1) * BUF_DW + lane * LROW_DW;
#pragma unroll
            for (int t = 0; t < TC; ++t) {            // fully unrolled: pad math is imm
                const int q = 3 * t;
                float x0 = lds[cb + q     + ( q      >> 5)];
                float x1 = lds[cb + q + 1 + ((q + 1) >> 5)];
                float x2 = lds[cb + q + 2 + ((q + 2) >> 5)];
                lstm_step(x0, x1, x2, wih, whh, bias, s);
            }
        }
    }

    // Tail timesteps and ragged (B%32) blocks: direct global path.
    if (b < B) {
        for (int t = chunks * TC; t < T; ++t) {
            const uint64_t o = ((uint64_t)b * (uint64_t)T + (uint64_t)t) * 3ull;
            lstm_step(story[o], story[o + 1], story[o + 2], wih, whh, bias, s);
        }
        // Cosine similarity vs both endings (eps = 1e-8, torch semantics).
        const float hh = s.h0 * s.h0 + s.h1 * s.h1 + s.h2 * s.h2;
        const float nh = fmaxf(sqrtf(hh), 1e-8f);

        float a0 = e1[3 * b], a1 = e1[3 * b + 1], a2 = e1[3 * b + 2];
        float n1 = fmaxf(sqrtf(a0 * a0 + a1 * a1 + a2 * a2), 1e-8f);
        float d1 = s.h0 * a0 + s.h1 * a1 + s.h2 * a2;

        float c0 = e2[3 * b], c1 = e2[3 * b + 1], c2 = e2[3 * b + 2];
        float n2 = fmaxf(sqrtf(c0 * c0 + c1 * c1 + c2 * c2), 1e-8f);
        float d2 = s.h0 * c0 + s.h1 * c1 + s.h2 * c2;

        out[2 * b + 0] = d1 * fast_rcp(nh * n1);
        out[2 * b + 1] = d2 * fast_rcp(nh * n2);
    }
}

extern "C" void kernel_launch(void* const* d_in, const int* in_sizes, int n_in,
                              void* d_out, int out_size, void* d_ws, size_t ws_size,
                              hipStream_t stream) {
    const float* story = (const float*)d_in[0];
    const float* e1    = (const float*)d_in[1];
    const float* e2    = (const float*)d_in[2];
    const float* Wih   = (const float*)d_in[3];
    const float* Whh   = (const float*)d_in[4];
    const float* bih   = (const float*)d_in[5];
    const float* bhh   = (const float*)d_in[6];
    float* out = (float*)d_out;

    const int B = in_sizes[1] / 3;                 // ending1_emb is [B,3]
    const int T = in_sizes[0] / (B * 3);           // story_emb is [B,T,3]
    const int blocks = (B + 31) / 32;

    lstm_cosine_kernel<<<blocks, 32, 0, stream>>>(story, e1, e2, Wih, Whh,
                                                  bih, bhh, out, B, T);
}